// my_lstm_14542759264270
// MI455X (gfx1250) — compile-verified
//
#include <hip/hip_runtime.h>

// ---------------------------------------------------------------------------
// LSTM on MI455X (gfx1250): bf16 WMMA, fp32 accumulate, async-LDS staging.
//   T=128, B=128, I=512, H=2048, O=512
// Step GEMM fuses [x_t | h_{t-1}] @ [Wx ; Wh]  (K = 2560) per timestep.
// ---------------------------------------------------------------------------

typedef __attribute__((ext_vector_type(8)))  float        v8f;
typedef __attribute__((ext_vector_type(16))) __bf16       v16bf;
typedef __attribute__((ext_vector_type(4)))  unsigned int v4u;

union Frag {
    v16bf bf;
    v4u   u[2];
};

#define T_   128
#define B_   128
#define I_   512
#define H_   2048
#define O_   512
#define G4_  (4 * H_)          // 8192
#define K_   (I_ + H_)         // 2560
#define KT_  32                // K consumed per WMMA
#define NKI_ (K_ / KT_)        // 80 k-tiles in recurrence
#define NKF_ (H_ / KT_)        // 64 k-tiles in final FC
#define NTH_ (H_ / 16)         // 128 column tiles per gate

__device__ __forceinline__ unsigned short f2bf(float f) {
    unsigned u = __float_as_uint(f);
    unsigned r = u + 0x7FFFu + ((u >> 16) & 1u);   // round-to-nearest-even
    return (unsigned short)(r >> 16);
}

__device__ __forceinline__ float sigf(float x) {
    return 1.0f / (1.0f + __expf(-x));
}

__device__ __forceinline__ float fast_tanh(float x) {
#if __has_builtin(__builtin_amdgcn_tanhf)
    return __builtin_amdgcn_tanhf(x);      // v_tanh_f32
#else
    return tanhf(x);
#endif
}

// gfx1250 async copy: 16 B global -> LDS, tracked by ASYNCcnt.
__device__ __forceinline__ void async_ld16(unsigned lds_byte_off, const void* gaddr) {
    asm volatile("global_load_async_to_lds_b128 %0, %1, off"
                 :: "v"(lds_byte_off), "v"(gaddr)
                 : "memory");
}
#define WAIT_ASYNC0() asm volatile("s_wait_asynccnt 0x0" ::: "memory")

// 4 accumulate-in-place WMMAs (one shared B) as a single block:
// tied "+v" keeps D=C in the same VGPRs (no acc rotation movs); trailing
// v_nops satisfy the ISA WMMA->VALU hazard (4 coexec slots for bf16 WMMA).
__device__ __forceinline__ void wmma4_sharedB(v8f acc[4], const Frag af[4], const Frag& bfr) {
    asm volatile(
        "v_wmma_f32_16x16x32_bf16 %0, %4, %8, %0\n\t"
        "v_wmma_f32_16x16x32_bf16 %1, %5, %8, %1\n\t"
        "v_wmma_f32_16x16x32_bf16 %2, %6, %8, %2\n\t"
        "v_wmma_f32_16x16x32_bf16 %3, %7, %8, %3\n\t"
        "v_nop\n\tv_nop\n\tv_nop\n\tv_nop"
        : "+v"(acc[0]), "+v"(acc[1]), "+v"(acc[2]), "+v"(acc[3])
        : "v"(af[0].bf), "v"(af[1].bf), "v"(af[2].bf), "v"(af[3].bf), "v"(bfr.bf));
}

// 2x2 accumulate-in-place WMMAs (2 A frags x 2 B frags) as a single block.
__device__ __forceinline__ void wmma2x2(v8f& a00, v8f& a01, v8f& a10, v8f& a11,
                                        const Frag& af0, const Frag& af1,
                                        const Frag& b0,  const Frag& b1) {
    asm volatile(
        "v_wmma_f32_16x16x32_bf16 %0, %4, %6, %0\n\t"
        "v_wmma_f32_16x16x32_bf16 %1, %4, %7, %1\n\t"
        "v_wmma_f32_16x16x32_bf16 %2, %5, %6, %2\n\t"
        "v_wmma_f32_16x16x32_bf16 %3, %5, %7, %3\n\t"
        "v_nop\n\tv_nop\n\tv_nop\n\tv_nop"
        : "+v"(a00), "+v"(a01), "+v"(a10), "+v"(a11)
        : "v"(af0.bf), "v"(af1.bf), "v"(b0.bf), "v"(b1.bf));
}

// --------------------------- prep kernels ----------------------------------

__global__ void k_cvt_x(const float* __restrict__ x, unsigned short* __restrict__ xb, int n) {
    int i = blockIdx.x * blockDim.x + threadIdx.x;
    if (i < n) xb[i] = f2bf(x[i]);
}

// Pack [Wx;Wh] (K_=2560 x 8192, row-major fp32) into WMMA B-fragment layout:
//   Wp[((nt*NKI_ + kt)*32 + lane)*16 + e] = bf16( W[kt*32 + (lane>>4)*16 + e][nt*16 + (lane&15)] )
__global__ void k_pack_w(const float* __restrict__ Wx, const float* __restrict__ Wh,
                         unsigned short* __restrict__ Wp) {
    int tid  = blockIdx.x * blockDim.x + threadIdx.x;   // (nt*NKI_+kt)*32 + lane
    int lane = tid & 31;
    int frag = tid >> 5;
    int kt   = frag % NKI_;
    int nt   = frag / NKI_;
    if (nt >= G4_ / 16) return;
    int n     = nt * 16 + (lane & 15);
    int kbase = kt * 32 + (lane >> 4) * 16;
    union { unsigned short s[16]; v4u u[2]; } vals;
#pragma unroll
    for (int e = 0; e < 16; ++e) {
        int k = kbase + e;
        float v = (k < I_) ? Wx[(size_t)k * G4_ + n] : Wh[(size_t)(k - I_) * G4_ + n];
        vals.s[e] = f2bf(v);
    }
    v4u* dst = (v4u*)(Wp + (size_t)tid * 16);
    dst[0] = vals.u[0];
    dst[1] = vals.u[1];
}

// Pack fc_w (2048 x 512, row-major fp32) into B-fragment layout.
__global__ void k_pack_fc(const float* __restrict__ fcw, unsigned short* __restrict__ Fp) {
    int tid  = blockIdx.x * blockDim.x + threadIdx.x;
    int lane = tid & 31;
    int frag = tid >> 5;
    int kt   = frag % NKF_;
    int nt   = frag / NKF_;
    if (nt >= O_ / 16) return;
    int n     = nt * 16 + (lane & 15);
    int kbase = kt * 32 + (lane >> 4) * 16;
    union { unsigned short s[16]; v4u u[2]; } vals;
#pragma unroll
    for (int e = 0; e < 16; ++e)
        vals.s[e] = f2bf(fcw[(size_t)(kbase + e) * O_ + n]);
    v4u* dst = (v4u*)(Fp + (size_t)tid * 16);
    dst[0] = vals.u[0];
    dst[1] = vals.u[1];
}

// zero h0 slot (bf16), zero c (fp32), and bsum = bx + bh
__global__ void k_init(unsigned short* __restrict__ hs0, float* __restrict__ cbuf,
                       const float* __restrict__ bx, const float* __restrict__ bh,
                       float* __restrict__ bsum, int n) {
    int i = blockIdx.x * blockDim.x + threadIdx.x;
    if (i < n) { hs0[i] = 0; cbuf[i] = 0.0f; }
    if (i < G4_) bsum[i] = bx[i] + bh[i];
}

// --------------------------- recurrent step --------------------------------
// Grid: 256 WGs = 2 (64-row halves of the batch) x 128 (16-wide H column tiles).
// 128 threads = 4 waves, one gate per wave; wave tile 64x16 (4 WMMA C-tiles).
// A tiles are double-buffered in LDS via async global->LDS copies (ASYNCcnt).
__global__ void __launch_bounds__(128)
k_lstm_step(const unsigned short* __restrict__ xb,   // [T,B,I] bf16
            const unsigned short* __restrict__ Wp,   // packed [Wx;Wh] bf16
            const float* __restrict__ bsum,          // bx+bh [4H]
            unsigned short* __restrict__ hs,         // [(T+1),B,H] bf16 (slot 0 = h_{-1}=0)
            float* __restrict__ cbuf,                // [B,H] fp32 (in/out)
            float* __restrict__ hbuf,                // [B,H] fp32 (out)
            int t)
{
    __shared__ __align__(16) float Gsm[4 * 64 * 16];        // 16 KB gate staging
    unsigned short* Abuf = (unsigned short*)Gsm;            // 2 x 4 KB A double-buffer (aliased)

    const int tid  = threadIdx.x;
    const int lane = tid & 31;
    const int wave = tid >> 5;          // gate index 0..3 (i,f,o,c)
    const int mb   = blockIdx.x & 1;    // which 64-row half of the batch
    const int nb   = blockIdx.x >> 1;   // 0..127: column tile of H
    const int rb0  = mb * 64;

    const v8f vzero = {0.f, 0.f, 0.f, 0.f, 0.f, 0.f, 0.f, 0.f};
    v8f acc[4];
#pragma unroll
    for (int m = 0; m < 4; ++m) acc[m] = vzero;

    const unsigned short* Bbase = Wp + ((size_t)(wave * NTH_ + nb) * NKI_) * 512;

    // async staging: thread covers half of one A row (16 bf16 = 32 B)
    const int srow  = tid >> 1;         // 0..63 local row
    const int shalf = tid & 1;
    const unsigned short* xsrc = xb + ((size_t)t * B_ + rb0 + srow) * I_ + shalf * 16;
    const unsigned short* hsrc = hs + ((size_t)t * B_ + rb0 + srow) * H_ + shalf * 16;
    const unsigned abase = (unsigned)(size_t)(void*)Abuf;                 // LDS byte offset
    const unsigned asoff = abase + (unsigned)(srow * 64 + shalf * 32);

    auto stageTo = [&](int kk, unsigned dbuf) {       // dbuf: 0 or 4096 (byte offset)
        const int k0 = kk * KT_;
        const unsigned short* s = (k0 < I_) ? (xsrc + k0) : (hsrc + (k0 - I_));
        const unsigned d = asoff + dbuf;
        async_ld16(d,      s);
        async_ld16(d + 16, s + 8);
    };

    const int half = lane >> 4;
    const int rl   = lane & 15;

    auto computeAt = [&](int kk, unsigned sbuf) {     // sbuf: 0 or 4096 (byte offset)
        Frag bfr;
        const v4u* b4 = (const v4u*)(Bbase + (size_t)kk * 512 + lane * 16);
        bfr.u[0] = b4[0]; bfr.u[1] = b4[1];
        const v4u* a4 = (const v4u*)((const char*)Abuf + sbuf);
        Frag af[4];
#pragma unroll
        for (int m = 0; m < 4; ++m) {       // batch all LDS loads first
            const int row = m * 16 + rl;
            af[m].u[0] = a4[row * 4 + half];
            af[m].u[1] = a4[row * 4 + 2 + half];
        }
        wmma4_sharedB(acc, af, bfr);
    };

    // prologue: stage k-tile 0, make it visible
    stageTo(0, 0);
    WAIT_ASYNC0();
    __syncthreads();

    // single-instance steady-state body: runtime-toggled buffer selector,
    // unrolling disabled so the tied-acc asm block has exactly one home.
    unsigned sel = 0;
#pragma clang loop unroll(disable)
    for (int kk = 0; kk < NKI_ - 1; ++kk) {
        stageTo(kk + 1, sel ^ 4096u);                            // overlap with compute
        __builtin_prefetch(Bbase + (size_t)(kk + 1) * 512, 0, 3);
        computeAt(kk, sel);
        WAIT_ASYNC0();       // my stage of kk+1 landed
        __syncthreads();     // everyone's landed; buf(kk) reads done
        sel ^= 4096u;
    }
    computeAt(NKI_ - 1, sel);

    __syncthreads();   // done reading Abuf alias; Gsm region is free
    {   // spill this wave's [64 x 16] gate tile to LDS
        const int col   = lane & 15;
        const int rbase = (lane >> 4) * 8;
#pragma unroll
        for (int m = 0; m < 4; ++m)
#pragma unroll
            for (int i = 0; i < 8; ++i)
                Gsm[wave * 1024 + (m * 16 + rbase + i) * 16 + col] = acc[m][i];
    }
    __syncthreads();

    {   // gate combine: 1024 elements over 128 threads (8 each, contiguous)
        const int n0 = nb * 16;
#pragma unroll
        for (int j = 0; j < 8; ++j) {
            const int idx = tid * 8 + j;
            const int lr  = idx >> 4;          // local row
            const int col = idx & 15;
            const int b   = rb0 + lr;
            const int n   = n0 + col;
            float gi = Gsm[0 * 1024 + idx] + bsum[0 * H_ + n];
            float gf = Gsm[1 * 1024 + idx] + bsum[1 * H_ + n];
            float go = Gsm[2 * 1024 + idx] + bsum[2 * H_ + n];
            float gc = Gsm[3 * 1024 + idx] + bsum[3 * H_ + n];
            float cp = cbuf[(size_t)b * H_ + n];
            float nc = sigf(gf) * cp + sigf(gi) * fast_tanh(gc);
            float nh = sigf(go) * fast_tanh(nc);
            cbuf[(size_t)b * H_ + n] = nc;
            hbuf[(size_t)b * H_ + n] = nh;
            hs[((size_t)(t + 1) * B_ + b) * H_ + n] = f2bf(nh);
        }
    }
}

// --------------------------- final FC GEMM ---------------------------------
// out[16384 x 512] = hs[16384 x 2048] @ fc_w + fc_b
// Grid: 128 M-blocks x 8 N-blocks; 256 threads = 8 waves; wave tile 32x32.
// Async double-buffered A staging, last iteration peeled, unroll disabled.
__global__ void __launch_bounds__(256)
k_fc(const unsigned short* __restrict__ hs,   // [(T+1),B,H] bf16; rows shifted by B_
     const unsigned short* __restrict__ Fp,   // packed fc_w bf16
     const float* __restrict__ fcb,
     float* __restrict__ out)
{
    __shared__ __align__(16) unsigned short Asm[2 * 128 * 32];   // 2 x 8 KB

    const int tid  = threadIdx.x;
    const int lane = tid & 31;
    const int wave = tid >> 5;
    const int wm   = wave >> 1;          // 0..3: 32-row sub-block
    const int wn   = wave & 1;           // 0..1: 32-col sub-block
    const int bm   = blockIdx.x & 127;   // M block (128 rows)
    const int bn   = blockIdx.x >> 7;    // N block (64 cols)

    const v8f vzero = {0.f, 0.f, 0.f, 0.f, 0.f, 0.f, 0.f, 0.f};
    v8f acc00 = vzero, acc01 = vzero, acc10 = vzero, acc11 = vzero;

    // async staging: thread covers half of one A row (32 B)
    const int srow  = tid >> 1;          // 0..127 local row
    const int shalf = tid & 1;
    const unsigned short* asrc =
        hs + ((size_t)bm * 128 + srow + B_) * H_ + shalf * 16;   // +B_ = slot shift
    const unsigned abase = (unsigned)(size_t)(void*)Asm;
    const unsigned asoff = abase + (unsigned)(srow * 64 + shalf * 32);

    auto stageTo = [&](int kk, unsigned dbuf) {       // dbuf: 0 or 8192
        const unsigned short* s = asrc + kk * KT_;
        const unsigned d = asoff + dbuf;
        async_ld16(d,      s);
        async_ld16(d + 16, s + 8);
    };

    const int half = lane >> 4;
    const int rl   = lane & 15;

    auto computeAt = [&](int kk, unsigned sbuf) {
        Frag bfr[2];
#pragma unroll
        for (int nn = 0; nn < 2; ++nn) {
            const int nt = bn * 4 + wn * 2 + nn;
            const v4u* b4 = (const v4u*)(Fp + ((size_t)(nt * NKF_ + kk) * 32 + lane) * 16);
            bfr[nn].u[0] = b4[0]; bfr[nn].u[1] = b4[1];
        }
        const v4u* a4 = (const v4u*)((const char*)Asm + sbuf);
        Frag af[2];
#pragma unroll
        for (int mm = 0; mm < 2; ++mm) {
            const int row = (wm * 2 + mm) * 16 + rl;
            af[mm].u[0] = a4[row * 4 + half];
            af[mm].u[1] = a4[row * 4 + 2 + half];
        }
        wmma2x2(acc00, acc01, acc10, acc11, af[0], af[1], bfr[0], bfr[1]);
    };

    stageTo(0, 0);
    WAIT_ASYNC0();
    __syncthreads();

    unsigned sel = 0;
#pragma clang loop unroll(disable)
    for (int kk = 0; kk < NKF_ - 1; ++kk) {
        stageTo(kk + 1, sel ^ 8192u);
        __builtin_prefetch(Fp + ((size_t)((bn * 4 + wn * 2) * NKF_ + kk + 1) * 32) * 16, 0, 3);
        computeAt(kk, sel);
        WAIT_ASYNC0();
        __syncthreads();
        sel ^= 8192u;
    }
    computeAt(NKF_ - 1, sel);

    const int colb  = lane & 15;
    const int rbase = (lane >> 4) * 8;
    const v8f accs[2][2] = {{acc00, acc01}, {acc10, acc11}};
#pragma unroll
    for (int mm = 0; mm < 2; ++mm)
#pragma unroll
        for (int nn = 0; nn < 2; ++nn)
#pragma unroll
            for (int i = 0; i < 8; ++i) {
                const int grow = bm * 128 + (wm * 2 + mm) * 16 + rbase + i;
                const int col  = bn * 64 + (wn * 2 + nn) * 16 + colb;
                out[(size_t)grow * O_ + col] = accs[mm][nn][i] + fcb[col];
            }
}

// write final (h, c) after outputs
__global__ void k_out_hc(const float* __restrict__ hbuf, const float* __restrict__ cbuf,
                         float* __restrict__ out) {
    int i = blockIdx.x * blockDim.x + threadIdx.x;
    if (i < B_ * H_) {
        out[(size_t)T_ * B_ * O_ + i]            = hbuf[i];
        out[(size_t)T_ * B_ * O_ + B_ * H_ + i]  = cbuf[i];
    }
}

// --------------------------- host launcher ---------------------------------

extern "C" void kernel_launch(void* const* d_in, const int* in_sizes, int n_in,
                              void* d_out, int out_size, void* d_ws, size_t ws_size,
                              hipStream_t stream) {
    const float* x   = (const float*)d_in[0];
    const float* Wx  = (const float*)d_in[1];
    const float* Wh  = (const float*)d_in[2];
    const float* bx  = (const float*)d_in[3];
    const float* bh  = (const float*)d_in[4];
    const float* fcw = (const float*)d_in[5];
    const float* fcb = (const float*)d_in[6];
    float* out = (float*)d_out;

    char* w = (char*)d_ws;
    size_t off = 0;
    auto alloc = [&](size_t bytes) {
        void* p = w + off;
        off = (off + bytes + 255) & ~(size_t)255;
        return p;
    };
    unsigned short* xb   = (unsigned short*)alloc((size_t)T_ * B_ * I_ * 2);          // 16 MB
    unsigned short* Wp   = (unsigned short*)alloc((size_t)K_ * G4_ * 2);              // 40 MB
    unsigned short* Fp   = (unsigned short*)alloc((size_t)H_ * O_ * 2);               //  2 MB
    unsigned short* hsB  = (unsigned short*)alloc((size_t)(T_ + 1) * B_ * H_ * 2);    // 64.5 MB
    float*          cbuf = (float*)alloc((size_t)B_ * H_ * 4);                        //  1 MB
    float*          hbuf = (float*)alloc((size_t)B_ * H_ * 4);                        //  1 MB
    float*          bsum = (float*)alloc((size_t)G4_ * 4);                            // 32 KB
    (void)ws_size; (void)in_sizes; (void)n_in; (void)out_size;

    // prep: bf16 convert + fragment packing + state init
    {
        int n = T_ * B_ * I_;
        k_cvt_x<<<(n + 255) / 256, 256, 0, stream>>>(x, xb, n);
    }
    k_pack_w<<<(G4_ / 16) * NKI_ * 32 / 256, 256, 0, stream>>>(Wx, Wh, Wp);
    k_pack_fc<<<(O_ / 16) * NKF_ * 32 / 256, 256, 0, stream>>>(fcw, Fp);
    k_init<<<(B_ * H_ + 255) / 256, 256, 0, stream>>>(hsB, cbuf, bx, bh, bsum, B_ * H_);

    // sequential recurrence: one launch per timestep (grid-wide dependency)
    for (int t = 0; t < T_; ++t)
        k_lstm_step<<<NTH_ * 2, 128, 0, stream>>>(xb, Wp, bsum, hsB, cbuf, hbuf, t);

    // final projection straight into d_out, then (h, c)
    k_fc<<<128 * (O_ / 64), 256, 0, stream>>>(hsB, Fp, fcb, out);
    k_out_hc<<<(B_ * H_ + 255) / 256, 256, 0, stream>>>(hbuf, cbuf, out);
}